// ESPInjector_7370163880192
// MI455X (gfx1250) — compile-verified
//
#include <hip/hip_runtime.h>
#include <hip/hip_bf16.h>
#include <math.h>

// ---------------------------------------------------------------------------
// ESPInjector fused pipeline for MI455X (gfx1250, wave32)
//
// Kernel 1 (esp_field): pairwise Coulomb ESP + E-field, one wave32 per query
//   row, lane-strided over NM=16384 MM atoms, wave shuffle reduction.
// Kernel 2 (fused_mlp): all MLP/GEMM stages fused per 16-row M-tile using
//   V_WMMA_F32_16X16X4_F32 (full fp32 matrix cores), intermediates in LDS.
// ---------------------------------------------------------------------------

#define B_     4
#define NQ_    96
#define NM_    16384
#define CS_    128
#define CV_    64
#define CSPHI_ 32
#define HID_   64
#define NROW   (B_ * NQ_)          // 384 rows total
#define EPS_   1e-12f

typedef __attribute__((ext_vector_type(2))) float v2f;
typedef __attribute__((ext_vector_type(8))) float v8f;

static __device__ __forceinline__ float siluf(float x) {
    return x / (1.0f + __expf(-x));
}

// D = A(16x4,f32) * B(4x16,f32) + C(16x16,f32)   -- wave32 matrix op
static __device__ __forceinline__ v8f wmma_f32_k4(v2f a, v2f b, v8f c) {
    return __builtin_amdgcn_wmma_f32_16x16x4_f32(
        /*neg_a=*/false, a, /*neg_b=*/false, b,
        /*c_mod=*/(short)0, c, /*reuse_a=*/false, /*reuse_b=*/false);
}

// Accumulate C += A[0:16, 0:K] @ W[0:K, n0:n0+16] over K (multiple of 4).
// A: row-major (lda), W: row-major (ldw).  A-frag: lane lo=m, hi selects K
// pair {k0+2hi, k0+2hi+1}.  B-frag mirrored with lane lo = column.
static __device__ __forceinline__ v8f gemm_tile(
    const float* __restrict__ A, int lda,
    const float* __restrict__ W, int ldw,
    int K, int n0, int lane, v8f acc)
{
    const int lo = lane & 15;
    const int hi = lane >> 4;
    const int n  = n0 + lo;
#pragma unroll 4
    for (int k0 = 0; k0 < K; k0 += 4) {
        v2f a, b;
        const float* pa = A + lo * lda + k0 + 2 * hi;
        a.x = pa[0];
        a.y = pa[1];
        const float* pb = W + (size_t)(k0 + 2 * hi) * ldw + n;
        b.x = pb[0];
        b.y = pb[ldw];
        acc = wmma_f32_k4(a, b, acc);
    }
    return acc;
}

// Broadcast bias[n] into all 8 accumulator rows.
static __device__ __forceinline__ v8f bias_acc(const float* __restrict__ bias,
                                               int n0, int lane) {
    float bv = bias[n0 + (lane & 15)];
    v8f c = {};
#pragma unroll
    for (int r = 0; r < 8; ++r) c[r] = bv;
    return c;
}

// Store C-frag (element (m = r + 8*hi, n = n0 + lo)) to row-major LDS buffer.
// act: 0 = none, 1 = SiLU, 2 = tanh
static __device__ __forceinline__ void store_acc_lds(float* buf, int ld, int n0,
                                                     int lane, v8f acc, int act) {
    const int lo = lane & 15;
    const int hi = lane >> 4;
#pragma unroll
    for (int r = 0; r < 8; ++r) {
        float v = acc[r];
        if (act == 1) v = siluf(v);
        else if (act == 2) v = tanhf(v);
        buf[(r + 8 * hi) * ld + n0 + lo] = v;
    }
}

// ---------------------------------------------------------------------------
// Kernel 1: ESP potential + E-field.  One wave per query row.
// ---------------------------------------------------------------------------
__global__ __launch_bounds__(256) void esp_field_kernel(
    const float* __restrict__ qm,          // (B,NQ,3)
    const float* __restrict__ mm,          // (B,NM,3)
    const float* __restrict__ qe,          // (B,NM)
    const unsigned char* __restrict__ rl,  // (B,NM) bool
    float* __restrict__ out_en,            // (NROW)
    float* __restrict__ out_ev,            // (NROW,3)
    float* __restrict__ ws_phi)            // (NROW)
{
    const int lane = threadIdx.x & 31;
    const int wave = threadIdx.x >> 5;
    const int row  = blockIdx.x * 8 + wave;      // 48 blocks * 8 waves = 384
    if (row >= NROW) return;                     // wave-uniform

    const int b = row / NQ_;
    const float qx = qm[row * 3 + 0];
    const float qy = qm[row * 3 + 1];
    const float qz = qm[row * 3 + 2];
    const float* mmb = mm + (size_t)b * NM_ * 3;
    const float* qb  = qe + (size_t)b * NM_;
    const unsigned char* rb = rl + (size_t)b * NM_;

    float phi = 0.0f, ex = 0.0f, ey = 0.0f, ez = 0.0f;
    for (int m = lane; m < NM_; m += 32) {
        float dx = qx - mmb[m * 3 + 0];
        float dy = qy - mmb[m * 3 + 1];
        float dz = qz - mmb[m * 3 + 2];
        float d2 = dx * dx + dy * dy + dz * dz;
        float d  = sqrtf(d2);
        float inv = rb[m] ? (1.0f / fmaxf(d, EPS_)) : 0.0f;
        float w   = qb[m] * inv;             // q * inv_d
        phi += w;
        float w3 = w * inv * inv;            // q * inv_d^3
        ex += w3 * dx;
        ey += w3 * dy;
        ez += w3 * dz;
    }
    // wave32 butterfly reduction
#pragma unroll
    for (int off = 16; off >= 1; off >>= 1) {
        phi += __shfl_xor(phi, off, 32);
        ex  += __shfl_xor(ex,  off, 32);
        ey  += __shfl_xor(ey,  off, 32);
        ez  += __shfl_xor(ez,  off, 32);
    }
    if (lane == 0) {
        const float FAC = 27.2114f * 0.529177249f;   // Hartree*Bohr -> eV*A
        phi *= FAC; ex *= FAC; ey *= FAC; ez *= FAC;
        ws_phi[row] = phi;
        out_ev[row * 3 + 0] = ex;
        out_ev[row * 3 + 1] = ey;
        out_ev[row * 3 + 2] = ez;
        float en = sqrtf(ex * ex + ey * ey + ez * ez);
        out_en[row] = fmaxf(en, EPS_);
    }
}

// ---------------------------------------------------------------------------
// Kernel 2: fused MLP pipeline.  One block (4 waves / 128 thr) per 16-row tile.
// ---------------------------------------------------------------------------
__global__ __launch_bounds__(128) void fused_mlp_kernel(
    const float* __restrict__ h_s0,   // (NROW,128)
    const float* __restrict__ h_v0,   // (NROW,3,64)
    const float* __restrict__ ws_phi, // (NROW)
    const float* __restrict__ en,     // (NROW)     (written by kernel 1)
    const float* __restrict__ ev,     // (NROW,3)   (written by kernel 1)
    const float* __restrict__ phi_w0, const float* __restrict__ phi_b0,
    const float* __restrict__ phi_w1, const float* __restrict__ phi_b1,
    const float* __restrict__ phi_w2, const float* __restrict__ phi_b2,
    const float* __restrict__ sproj_w, const float* __restrict__ sproj_b,
    const float* __restrict__ eg_w0, const float* __restrict__ eg_b0,
    const float* __restrict__ eg_w1, const float* __restrict__ eg_b1,
    const float* __restrict__ eg_w2, const float* __restrict__ eg_b2,
    float* __restrict__ out_hs,       // (NROW,128)
    float* __restrict__ out_hv,       // (NROW,3,64)
    float* __restrict__ out_pf)       // (NROW,32)
{
    __shared__ float s_hs0[16 * CS_];     // h_s0 tile
    __shared__ float s_t0[16 * HID_];     // phi MLP hidden 0
    __shared__ float s_t1[16 * HID_];     // phi MLP hidden 1
    __shared__ float s_pf[16 * CSPHI_];   // phi_feat
    __shared__ float s_hs[16 * CS_];      // h_s
    __shared__ float s_a1[16 * HID_];     // E-gate hidden 0
    __shared__ float s_a2[16 * HID_];     // E-gate hidden 1
    __shared__ float s_g [16 * CV_];      // gamma
    __shared__ float s_phi[16];
    __shared__ float s_en[16];
    __shared__ float s_ev[16 * 3];

    const int t    = threadIdx.x;
    const int lane = t & 31;
    const int wave = t >> 5;
    const int lo   = lane & 15;
    const int hi   = lane >> 4;
    const int r0   = blockIdx.x * 16;     // first row of this tile

    // ---- stage in tile inputs -------------------------------------------
    for (int i = t; i < 16 * CS_; i += 128)
        s_hs0[i] = h_s0[(size_t)r0 * CS_ + i];
    if (t < 16) {
        s_phi[t] = ws_phi[r0 + t];
        s_en[t]  = en[r0 + t];
    }
    if (t < 48)
        s_ev[t] = ev[(size_t)r0 * 3 + t];
    __syncthreads();

    // ---- phi MLP layer 0 : (16,1)->(16,64), SiLU (pure VALU, din=1) -----
    for (int i = t; i < 16 * HID_; i += 128) {
        int m = i >> 6, j = i & 63;
        s_t0[i] = siluf(s_phi[m] * phi_w0[j] + phi_b0[j]);
    }
    __syncthreads();

    // ---- phi MLP layer 1 : (16,64)@(64,64), SiLU -- WMMA, 1 N-tile/wave --
    {
        int n0 = wave * 16;
        v8f acc = bias_acc(phi_b1, n0, lane);
        acc = gemm_tile(s_t0, HID_, phi_w1, HID_, HID_, n0, lane, acc);
        store_acc_lds(s_t1, HID_, n0, lane, acc, /*silu*/1);
    }
    __syncthreads();

    // ---- phi MLP layer 2 : (16,64)@(64,32) -- waves 0,1 ------------------
    if (wave < 2) {
        int n0 = wave * 16;
        v8f acc = bias_acc(phi_b2, n0, lane);
        acc = gemm_tile(s_t1, HID_, phi_w2, CSPHI_, HID_, n0, lane, acc);
        store_acc_lds(s_pf, CSPHI_, n0, lane, acc, 0);
#pragma unroll
        for (int r = 0; r < 8; ++r)
            out_pf[(size_t)(r0 + r + 8 * hi) * CSPHI_ + n0 + lo] = acc[r];
    }
    __syncthreads();

    // ---- s_proj : concat(h_s0,phi_feat,Enorm)(16,161)@(161,128) ----------
    for (int nt = wave; nt < 8; nt += 4) {
        int n0 = nt * 16;
        v8f acc = bias_acc(sproj_b, n0, lane);
        // K = 0..127 : h_s0 rows of the weight
        acc = gemm_tile(s_hs0, CS_, sproj_w, CS_, CS_, n0, lane, acc);
        // K = 128..159 : phi_feat rows of the weight
        acc = gemm_tile(s_pf, CSPHI_, sproj_w + 128 * CS_, CS_, CSPHI_, n0, lane, acc);
        // K = 160 : Enorm column, rank-1 VALU update on the C fragment
        float wrow = sproj_w[160 * CS_ + n0 + lo];
#pragma unroll
        for (int r = 0; r < 8; ++r) acc[r] += s_en[r + 8 * hi] * wrow;
        store_acc_lds(s_hs, CS_, n0, lane, acc, 0);
#pragma unroll
        for (int r = 0; r < 8; ++r)
            out_hs[(size_t)(r0 + r + 8 * hi) * CS_ + n0 + lo] = acc[r];
    }
    __syncthreads();

    // ---- E-gate layer 0 : concat(h_s,Enorm)(16,129)@(129,64), SiLU -------
    {
        int n0 = wave * 16;
        v8f acc = bias_acc(eg_b0, n0, lane);
        acc = gemm_tile(s_hs, CS_, eg_w0, HID_, CS_, n0, lane, acc);
        float wrow = eg_w0[128 * HID_ + n0 + lo];     // Enorm rank-1 term
#pragma unroll
        for (int r = 0; r < 8; ++r) acc[r] += s_en[r + 8 * hi] * wrow;
        store_acc_lds(s_a1, HID_, n0, lane, acc, /*silu*/1);
    }
    __syncthreads();

    // ---- E-gate layer 1 : (16,64)@(64,64), SiLU --------------------------
    {
        int n0 = wave * 16;
        v8f acc = bias_acc(eg_b1, n0, lane);
        acc = gemm_tile(s_a1, HID_, eg_w1, HID_, HID_, n0, lane, acc);
        store_acc_lds(s_a2, HID_, n0, lane, acc, /*silu*/1);
    }
    __syncthreads();

    // ---- E-gate layer 2 : (16,64)@(64,64), tanh -> gamma -----------------
    {
        int n0 = wave * 16;
        v8f acc = bias_acc(eg_b2, n0, lane);
        acc = gemm_tile(s_a2, HID_, eg_w2, CV_, HID_, n0, lane, acc);
        store_acc_lds(s_g, CV_, n0, lane, acc, /*tanh*/2);
    }
    __syncthreads();

    // ---- h_v = h_v0 + gamma[:,None,:] * Evec[:,:,None] -------------------
    for (int i = t; i < 16 * 3 * CV_; i += 128) {
        int m   = i / (3 * CV_);
        int rem = i - m * (3 * CV_);
        int d   = rem >> 6;
        int c   = rem & 63;
        size_t gidx = (size_t)(r0 + m) * (3 * CV_) + rem;
        out_hv[gidx] = h_v0[gidx] + s_g[m * CV_ + c] * s_ev[m * 3 + d];
    }
}

// ---------------------------------------------------------------------------
extern "C" void kernel_launch(void* const* d_in, const int* in_sizes, int n_in,
                              void* d_out, int out_size, void* d_ws, size_t ws_size,
                              hipStream_t stream) {
    const float* h_s0   = (const float*)d_in[0];
    const float* h_v0   = (const float*)d_in[1];
    const float* qm     = (const float*)d_in[2];
    const float* mm     = (const float*)d_in[3];
    const float* q_eff  = (const float*)d_in[4];
    const unsigned char* real_mm = (const unsigned char*)d_in[5];
    const float* phi_w0 = (const float*)d_in[6];
    const float* phi_b0 = (const float*)d_in[7];
    const float* phi_w1 = (const float*)d_in[8];
    const float* phi_b1 = (const float*)d_in[9];
    const float* phi_w2 = (const float*)d_in[10];
    const float* phi_b2 = (const float*)d_in[11];
    const float* sproj_w = (const float*)d_in[12];
    const float* sproj_b = (const float*)d_in[13];
    const float* eg_w0  = (const float*)d_in[14];
    const float* eg_b0  = (const float*)d_in[15];
    const float* eg_w1  = (const float*)d_in[16];
    const float* eg_b1  = (const float*)d_in[17];
    const float* eg_w2  = (const float*)d_in[18];
    const float* eg_b2  = (const float*)d_in[19];

    // Output layout (flat f32, reference return order):
    float* out    = (float*)d_out;
    float* out_hs = out;                              // (384,128)
    float* out_hv = out + NROW * CS_;                 // (384,3,64)
    float* out_pf = out_hv + NROW * 3 * CV_;          // (384,32)
    float* out_en = out_pf + NROW * CSPHI_;           // (384)
    float* out_ev = out_en + NROW;                    // (384,3)

    float* ws_phi = (float*)d_ws;                     // (384) scratch

    esp_field_kernel<<<NROW / 8, 256, 0, stream>>>(
        qm, mm, q_eff, real_mm, out_en, out_ev, ws_phi);

    fused_mlp_kernel<<<NROW / 16, 128, 0, stream>>>(
        h_s0, h_v0, ws_phi, out_en, out_ev,
        phi_w0, phi_b0, phi_w1, phi_b1, phi_w2, phi_b2,
        sproj_w, sproj_b,
        eg_w0, eg_b0, eg_w1, eg_b1, eg_w2, eg_b2,
        out_hs, out_hv, out_pf);
}